// GCNRegressor_65970697667357
// MI455X (gfx1250) — compile-verified
//
#include <hip/hip_runtime.h>
#include <hip/hip_bf16.h>
#include <cstddef>

// ---------------------------------------------------------------------------
// GCN (3-layer) for MI455X / gfx1250.
// GEMMs use V_WMMA_F32_16X16X4_F32 (exact fp32 matrix path, wave32).
// W is staged in LDS pre-packed into B-fragment layout so the inner loop does
// one conflict-free ds_load_b64 per fragment (no repacking moves).
// ---------------------------------------------------------------------------

typedef float v2f __attribute__((ext_vector_type(2)));
typedef float v8f __attribute__((ext_vector_type(8)));

#define D_FEAT 128
#define EPS_BN 1e-5f
#define LRELU  0.01f

// ---------------- degree kernels ----------------
__global__ void k_deg_init(float* __restrict__ deg, int N) {
    int i = blockIdx.x * blockDim.x + threadIdx.x;
    if (i < N) deg[i] = 1.0f;                 // self-loop
}

__global__ void k_deg_count(const int* __restrict__ dst, float* __restrict__ deg, int E) {
    int e = blockIdx.x * blockDim.x + threadIdx.x;
    if (e < E) atomicAdd(&deg[dst[e]], 1.0f);
}

__global__ void k_deg_fin(const float* __restrict__ deg,
                          float* __restrict__ dinv, float* __restrict__ selfc, int N) {
    int i = blockIdx.x * blockDim.x + threadIdx.x;
    if (i < N) {
        float d = deg[i];
        dinv[i]  = rsqrtf(d);
        selfc[i] = 1.0f / d;
    }
}

__global__ void k_zero(float* __restrict__ p, int n) {
    int i = blockIdx.x * blockDim.x + threadIdx.x;
    if (i < n) p[i] = 0.0f;
}

// ---------------- WMMA fp32 GEMM: Y[N,128] = X[N,128] @ W[128,128] ----------
// Block = 256 threads = 8 waves. Block covers 128 rows; wave w owns the
// 16-row tile (blockIdx.x*8 + w) across all 8 column tiles.
// LDS holds W pre-packed as B fragments: Wfrag[(ks*8 + t)*32 + lane] =
//   { W[ks*4 + 2*(lane>>4)][t*16 + (lane&15)],
//     W[ks*4 + 2*(lane>>4) + 1][t*16 + (lane&15)] }
__global__ __launch_bounds__(256) void k_gemm128_wmma(
    const float* __restrict__ X, const float* __restrict__ W,
    float* __restrict__ Y, int N)
{
    __shared__ v2f Wfrag[32 * 8 * 32];   // 64 KB, exactly one fragment per elem pair

    const int tid = threadIdx.x;
    // Stage W (128x128 f32) into B-fragment layout.
    for (int f = tid; f < 32 * 8 * 32; f += 256) {
        int flane = f & 31;
        int t     = (f >> 5) & 7;
        int ks    = f >> 8;
        int row   = ks * 4 + 2 * (flane >> 4);
        int col   = t * 16 + (flane & 15);
        v2f w;
        w.x = W[(size_t)row * D_FEAT + col];
        w.y = W[(size_t)(row + 1) * D_FEAT + col];
        Wfrag[f] = w;
    }
    __syncthreads();

    const int wave = tid >> 5;
    const int lane = tid & 31;
    const int half = lane >> 4;   // 0: lanes 0-15, 1: lanes 16-31
    const int lm   = lane & 15;

    const int rowTile = blockIdx.x * 8 + wave;     // 16-row tile index
    int row  = rowTile * 16 + lm;
    int rowC = (row < N) ? row : (N - 1);          // clamp loads; stores guarded
    const float* xrow = X + (size_t)rowC * D_FEAT;

    v8f acc[8];
#pragma unroll
    for (int t = 0; t < 8; ++t) acc[t] = (v8f)0.0f;

    // K loop: 128 in steps of 4 (V_WMMA_F32_16X16X4_F32)
    for (int ks = 0; ks < 32; ++ks) {
        const int k = ks * 4;
        // A fragment (16x4): lane<16 holds K = {k,k+1}, lane>=16 holds {k+2,k+3}
        v2f a = *(const v2f*)(xrow + k + 2 * half);
        const v2f* bline = &Wfrag[ks * 256 + lane];
#pragma unroll
        for (int t = 0; t < 8; ++t) {
            v2f b = bline[t * 32];   // one ds_load_b64, conflict-free
            acc[t] = __builtin_amdgcn_wmma_f32_16x16x4_f32(
                false, a, false, b, (short)0, acc[t], false, false);
        }
    }

    // C/D layout: VGPR r -> tile row (r + 8*half), tile col = lm
#pragma unroll
    for (int r = 0; r < 8; ++r) {
        int orow = rowTile * 16 + r + 8 * half;
        if (orow < N) {
#pragma unroll
            for (int t = 0; t < 8; ++t)
                Y[(size_t)orow * D_FEAT + t * 16 + lm] = acc[t][r];
        }
    }
}

// ---------------- aggregation ----------------
// AGG[n,f] = XW[n,f] * selfc[n] + b[f]   (self-loop term + bias, full overwrite)
__global__ void k_agg_init(const float* __restrict__ XW, const float* __restrict__ selfc,
                           const float* __restrict__ b, float* __restrict__ AGG, int total)
{
    int i = blockIdx.x * blockDim.x + threadIdx.x;
    if (i >= total) return;
    int n = i >> 7, f = i & (D_FEAT - 1);
    AGG[i] = XW[i] * selfc[n] + b[f];
}

// One wave per edge; lane moves a float4 of the 128-float message.
__global__ __launch_bounds__(256) void k_edge_scatter(
    const float* __restrict__ XW, const int* __restrict__ src, const int* __restrict__ dst,
    const float* __restrict__ dinv, float* __restrict__ AGG, int E)
{
    int gid  = blockIdx.x * blockDim.x + threadIdx.x;
    int e    = gid >> 5;
    int lane = threadIdx.x & 31;
    if (e >= E) return;
    int s = src[e], d = dst[e];
    float c = dinv[s] * dinv[d];
    float4 v = ((const float4*)(XW + (size_t)s * D_FEAT))[lane];
    float* out = AGG + (size_t)d * D_FEAT + lane * 4;
    atomicAdd(out + 0, v.x * c);
    atomicAdd(out + 1, v.y * c);
    atomicAdd(out + 2, v.z * c);
    atomicAdd(out + 3, v.w * c);
}

// ---------------- batch norm ----------------
__global__ __launch_bounds__(128) void k_bn_stats(
    const float* __restrict__ H, float* __restrict__ bsum, float* __restrict__ bsq, int N)
{
    int f = threadIdx.x;
    float s = 0.f, q = 0.f;
    for (int r = blockIdx.x; r < N; r += gridDim.x) {
        float v = H[(size_t)r * D_FEAT + f];
        s += v; q += v * v;
    }
    atomicAdd(&bsum[f], s);
    atomicAdd(&bsq[f],  q);
}

__global__ __launch_bounds__(128) void k_bn_finalize(
    const float* __restrict__ bsum, const float* __restrict__ bsq,
    float* __restrict__ mu, float* __restrict__ rstd, int N)
{
    int f = threadIdx.x;
    float inv_n = 1.0f / (float)N;
    float m = bsum[f] * inv_n;
    float var = bsq[f] * inv_n - m * m;
    mu[f]   = m;
    rstd[f] = rsqrtf(var + EPS_BN);
}

__global__ void k_bn_apply_lrelu(
    const float* __restrict__ Hin, const float* __restrict__ mu, const float* __restrict__ rstd,
    const float* __restrict__ g, const float* __restrict__ beta,
    float* __restrict__ Hout, int total)
{
    int i = blockIdx.x * blockDim.x + threadIdx.x;
    if (i >= total) return;
    int f = i & (D_FEAT - 1);
    float y = (Hin[i] - mu[f]) * rstd[f] * g[f] + beta[f];
    Hout[i] = (y >= 0.f) ? y : LRELU * y;
}

// ---------------- layer 3 (128 -> 1) ----------------
__global__ __launch_bounds__(256) void k_matvec128(
    const float* __restrict__ H, const float* __restrict__ W3, float* __restrict__ Y, int N)
{
    int gid  = blockIdx.x * blockDim.x + threadIdx.x;
    int n    = gid >> 5;
    int lane = threadIdx.x & 31;
    if (n >= N) return;
    float4 h = ((const float4*)(H + (size_t)n * D_FEAT))[lane];
    float4 w = ((const float4*)W3)[lane];
    float p = h.x * w.x + h.y * w.y + h.z * w.z + h.w * w.w;
#pragma unroll
    for (int off = 16; off > 0; off >>= 1)
        p += __shfl_xor(p, off, 32);
    if (lane == 0) Y[n] = p;
}

__global__ void k_out_init(const float* __restrict__ XW3, const float* __restrict__ selfc,
                           const float* __restrict__ b3, float* __restrict__ out, int N)
{
    int n = blockIdx.x * blockDim.x + threadIdx.x;
    if (n < N) out[n] = XW3[n] * selfc[n] + b3[0];
}

__global__ void k_out_edges(const float* __restrict__ XW3,
                            const int* __restrict__ src, const int* __restrict__ dst,
                            const float* __restrict__ dinv, float* __restrict__ out, int E)
{
    int e = blockIdx.x * blockDim.x + threadIdx.x;
    if (e >= E) return;
    int s = src[e], d = dst[e];
    atomicAdd(&out[d], XW3[s] * dinv[s] * dinv[d]);
}

// ---------------------------------------------------------------------------
extern "C" void kernel_launch(void* const* d_in, const int* in_sizes, int n_in,
                              void* d_out, int out_size, void* d_ws, size_t ws_size,
                              hipStream_t stream)
{
    const float* x     = (const float*)d_in[0];
    const int*   eidx  = (const int*)  d_in[1];
    const float* W1    = (const float*)d_in[2];
    const float* b1    = (const float*)d_in[3];
    const float* g1    = (const float*)d_in[4];
    const float* beta1 = (const float*)d_in[5];
    const float* W2    = (const float*)d_in[6];
    const float* b2    = (const float*)d_in[7];
    const float* g2    = (const float*)d_in[8];
    const float* beta2 = (const float*)d_in[9];
    const float* W3    = (const float*)d_in[10];
    const float* b3    = (const float*)d_in[11];

    const int N = in_sizes[0] / D_FEAT;   // 50000
    const int E = in_sizes[1] / 2;        // 800000
    const int* src = eidx;
    const int* dst = eidx + E;
    const int total = N * D_FEAT;

    // workspace carve-up (floats)
    float* ws    = (float*)d_ws;
    float* dinv  = ws;                    // N
    float* selfc = dinv + N;              // N
    float* degb  = selfc + N;             // N (temp)
    float* bufB  = degb + N;              // N*128
    float* bufC  = bufB + (size_t)N * D_FEAT;  // N*128
    float* xw3   = bufC + (size_t)N * D_FEAT;  // N
    float* bsum  = xw3 + N;               // 128
    float* bsq   = bsum + D_FEAT;         // 128 (contiguous with bsum)
    float* bmu   = bsq + D_FEAT;          // 128
    float* brstd = bmu + D_FEAT;          // 128

    float* out = (float*)d_out;

    const int T = 256;
    dim3 blkN((N + T - 1) / T);
    dim3 blkE((E + T - 1) / T);
    dim3 blkTot((total + T - 1) / T);
    dim3 blkGemm((N + 127) / 128);        // 128 rows per block
    dim3 blkEdgeWave(((size_t)E * 32 + T - 1) / T);
    dim3 blkNodeWave(((size_t)N * 32 + T - 1) / T);

    // ---- degrees & symmetric norm coefficients ----
    k_deg_init<<<blkN, T, 0, stream>>>(degb, N);
    k_deg_count<<<blkE, T, 0, stream>>>(dst, degb, E);
    k_deg_fin<<<blkN, T, 0, stream>>>(degb, dinv, selfc, N);

    // ---- layer 1: conv -> BN -> leaky_relu ----
    k_gemm128_wmma<<<blkGemm, T, 0, stream>>>(x, W1, bufB, N);          // xw1 -> B
    k_agg_init<<<blkTot, T, 0, stream>>>(bufB, selfc, b1, bufC, total); // agg1 -> C
    k_edge_scatter<<<blkEdgeWave, T, 0, stream>>>(bufB, src, dst, dinv, bufC, E);
    k_zero<<<1, 256, 0, stream>>>(bsum, 2 * D_FEAT);
    k_bn_stats<<<512, D_FEAT, 0, stream>>>(bufC, bsum, bsq, N);
    k_bn_finalize<<<1, D_FEAT, 0, stream>>>(bsum, bsq, bmu, brstd, N);
    k_bn_apply_lrelu<<<blkTot, T, 0, stream>>>(bufC, bmu, brstd, g1, beta1, bufB, total); // h1 -> B

    // ---- layer 2: conv -> BN -> leaky_relu ----
    k_gemm128_wmma<<<blkGemm, T, 0, stream>>>(bufB, W2, bufC, N);       // xw2 -> C
    k_agg_init<<<blkTot, T, 0, stream>>>(bufC, selfc, b2, bufB, total); // agg2 -> B
    k_edge_scatter<<<blkEdgeWave, T, 0, stream>>>(bufC, src, dst, dinv, bufB, E);
    k_zero<<<1, 256, 0, stream>>>(bsum, 2 * D_FEAT);
    k_bn_stats<<<512, D_FEAT, 0, stream>>>(bufB, bsum, bsq, N);
    k_bn_finalize<<<1, D_FEAT, 0, stream>>>(bsum, bsq, bmu, brstd, N);
    k_bn_apply_lrelu<<<blkTot, T, 0, stream>>>(bufB, bmu, brstd, g2, beta2, bufC, total); // h2 -> C

    // ---- layer 3: conv (128 -> 1) ----
    k_matvec128<<<blkNodeWave, T, 0, stream>>>(bufC, W3, xw3, N);
    k_out_init<<<blkN, T, 0, stream>>>(xw3, selfc, b3, out, N);
    k_out_edges<<<blkE, T, 0, stream>>>(xw3, src, dst, dinv, out, E);
}